// BlurDownsample_1726576853859
// MI455X (gfx1250) — compile-verified
//
#include <hip/hip_runtime.h>

// BlurDownsample (upfirdn2d, up=1, down=2, 4x4 separable filter) for MI455X.
// x: [16,128,256,256] f32 -> y: [16,128,128,128] f32.
// HBM-bound (AI ~1.6 FLOP/B, ~670MB => ~29us at 23.3 TB/s):
//  - stage input tiles with GLOBAL_LOAD_ASYNC_TO_LDS_B128 (ASYNCcnt path)
//  - separable filter: horizontal pass in VALU from LDS
//  - vertical stride-2 pass as banded-matrix v_wmma_f32_16x16x4_f32 (9 chunks)

typedef __attribute__((ext_vector_type(2))) float v2f;
typedef __attribute__((ext_vector_type(4))) float v4f;
typedef __attribute__((ext_vector_type(8))) float v8f;

#define H_IN   256
#define W_IN   256
#define H_OUT  128
#define W_OUT  128

// Select tap d (0..3) with a cndmask chain (no scratch spills).
__device__ __forceinline__ float tap_sel(int d, float t0, float t1, float t2, float t3) {
    float r = 0.0f;
    r = (d == 0) ? t0 : r;
    r = (d == 1) ? t1 : r;
    r = (d == 2) ? t2 : r;
    r = (d == 3) ? t3 : r;
    return r;
}

__global__ __launch_bounds__(256) void blurdown_wmma_kernel(
    const float* __restrict__ x,   // [planes, 256, 256]
    const float* __restrict__ f,   // [4,4] normalized separable filter
    float* __restrict__ y,         // [planes, 128, 128]
    int planes) {

    // Input staging tile: rows i=0..33 map to global rows 32*tile_row-1+i.
    // Column layout: LDS col l = gc + 4 (gc = global col, -1..256 -> l = 3..260).
    // Interior (gc 0..255) lives at l=4..259: byte offset 16 + i*1056, both
    // 16B-aligned so async b128 memory->LDS copies are legal. Row stride
    // 264 dwords (1056B = 66*16B).
    __shared__ float xs[34][264];          // 35,904 B
    __shared__ float hs[8][36][16];        // per-wave H panel, K padded to 36

    const int tid  = threadIdx.x;
    const int lane = tid & 31;
    const int w    = tid >> 5;             // wave id 0..7 -> output cols 16w..16w+15

    const int tiles_per_plane = H_OUT / 16;          // 8
    const int tile_row = blockIdx.x % tiles_per_plane;
    const int plane    = blockIdx.x / tiles_per_plane;
    if (plane >= planes) return;

    const float* __restrict__ xp = x + (size_t)plane * (H_IN * W_IN);
    float*       __restrict__ yp = y + (size_t)plane * (H_OUT * W_OUT);

    // Derive 1-D taps from the 4x4 filter (f = outer(t,t), t = row0/sqrt(f00)),
    // applied in flipped order (reference does a true convolution).
    const float s   = 1.0f / __builtin_sqrtf(f[0]);
    const float th0 = f[3]  * s, th1 = f[2]  * s, th2 = f[1] * s, th3 = f[0] * s;  // horiz
    const float tv0 = f[12] * s, tv1 = f[8]  * s, tv2 = f[4] * s, tv3 = f[0] * s;  // vert

    // --- Stage 0: global -> LDS via async-tensor path.
    // Interior: 34 rows x 64 chunks of 16B each (256 floats/row).
    const int row_in_base = 32 * tile_row - 1;
    for (int q = tid; q < 34 * 64; q += 256) {
        const int i   = q >> 6;            // tile row 0..33
        const int c16 = q & 63;            // 16B chunk within row
        const int gr  = row_in_base + i;
        float* lp = &xs[i][4 + 4 * c16];
        if (gr >= 0 && gr < H_IN) {
            const float* gp = xp + (gr * W_IN + 4 * c16);
            // memory -> LDS, no VGPR round-trip; tracked by ASYNCcnt.
            asm volatile("global_load_async_to_lds_b128 %0, %1, off"
                         :: "v"((unsigned)(uintptr_t)lp),  // AS(3) addr = low 32 bits
                            "v"(gp)
                         : "memory");
        } else {
            *reinterpret_cast<v4f*>(lp) = (v4f){};         // zero-pad OOB rows
        }
    }
    // Zero-padding columns gc=-1 (l=3) and gc=256 (l=260): always zero.
    if (tid < 68) xs[tid >> 1][(tid & 1) ? 260 : 3] = 0.0f;

    asm volatile("s_wait_asynccnt 0x0" ::: "memory");       // own copies landed
    __syncthreads();                                        // visible to all waves

    // --- Stage 1 (horizontal, VALU from LDS):
    // H[i][J] = sum_b thf[b] * xs[i][32w + 2J + 3 + b], i=0..33, J=0..15.
    #pragma unroll
    for (int t = 0; t < 17; ++t) {
        const int h  = lane + 32 * t;      // 0..543
        const int i  = h >> 4;
        const int J  = h & 15;
        const int l0 = 32 * w + 2 * J + 3;
        const float acc = th0 * xs[i][l0]     + th1 * xs[i][l0 + 1]
                        + th2 * xs[i][l0 + 2] + th3 * xs[i][l0 + 3];
        hs[w][i][J] = acc;
    }
    // Zero K-padding rows 34,35 (one entry per lane).
    hs[w][34 + (lane >> 4)][lane & 15] = 0.0f;
    __syncthreads();

    // --- Stage 2 (vertical, WMMA): Out[16x16] = A[16x36] x H[36x16],
    // A[r][i] = tvf[i-2r] for i-2r in [0,3], else 0 (stride-2 band matrix).
    // f32 16x16x4 layouts (ISA 7.12.2): A lane m=lane&15, VGPR0=K(k0), VGPR1=K(k0+1),
    // half-waves split K{0,1}/K{2,3}; B VGPR0 = H[k0][N], VGPR1 = H[k0+1][N].
    const int  m  = lane & 15;
    const bool hi = lane >= 16;

    v8f acc = {};
    #pragma unroll
    for (int c4 = 0; c4 < 9; ++c4) {
        const int k0 = 4 * c4 + (hi ? 2 : 0);
        const int d0 = k0 - 2 * m;
        v2f a, b;
        a.x = tap_sel(d0,     tv0, tv1, tv2, tv3);
        a.y = tap_sel(d0 + 1, tv0, tv1, tv2, tv3);
        b.x = hs[w][k0][m];
        b.y = hs[w][k0 + 1][m];
        acc = __builtin_amdgcn_wmma_f32_16x16x4_f32(
            /*neg_a=*/false, a, /*neg_b=*/false, b,
            /*c_mod=*/(short)0, acc, /*reuse_a=*/false, /*reuse_b=*/false);
    }

    // --- Store: C/D VGPR v -> out row v (lanes 0-15) / v+8 (lanes 16-31), col N.
    // Each half-wave writes 16 consecutive floats (64B segments) per VGPR.
    const int out_row_base = tile_row * 16 + (hi ? 8 : 0);
    const int out_col      = 16 * w + m;
    #pragma unroll
    for (int v = 0; v < 8; ++v) {
        yp[(out_row_base + v) * W_OUT + out_col] = acc[v];
    }
}

extern "C" void kernel_launch(void* const* d_in, const int* in_sizes, int n_in,
                              void* d_out, int out_size, void* d_ws, size_t ws_size,
                              hipStream_t stream) {
    const float* x = (const float*)d_in[0];
    const float* f = (const float*)d_in[1];
    float* y = (float*)d_out;

    const int planes = in_sizes[0] / (H_IN * W_IN);      // N*C = 2048
    const int tiles_per_plane = H_OUT / 16;              // 8
    dim3 grid(planes * tiles_per_plane);                 // 16384 workgroups
    dim3 block(256);                                     // 8 waves (wave32)

    blurdown_wmma_kernel<<<grid, block, 0, stream>>>(x, f, y, planes);
}